// CtrlGenModel_56135222558916
// MI455X (gfx1250) — compile-verified
//
#include <hip/hip_runtime.h>
#include <hip/hip_bf16.h>
#include <math.h>

// ---------------- model dimensions ----------------
static constexpr int cV  = 9000;   // vocab
static constexpr int cE  = 300;    // embedding
static constexpr int cH  = 700;    // hidden
static constexpr int cDC = 200;
static constexpr int cB  = 64;     // batch
static constexpr int cT  = 20;     // seq len
static constexpr int cS  = 19;     // encoder steps = T-1
static constexpr int cNF = 128;
static constexpr int c3H = 3 * cH; // 2100

typedef float v2f __attribute__((ext_vector_type(2)));
typedef float v8f __attribute__((ext_vector_type(8)));

// =====================================================================
// WMMA f32 GEMM:  C[M,N] = A[M,K] @ op(B) + bias
//   TRANS_B=true : B is weight stored [N,K] (row-major, ldb = row stride)
//   TRANS_B=false: B is [K,N] (row-major, ldb = row stride)
// One wave computes a 32x32 output tile: 2 A fragments x 2 B fragments
// -> 4 independent V_WMMA_F32_16X16X4_F32 accumulator chains per K-step
// (4 global_load_b64 : 4 wmma in the inner loop; each fragment reused 2x).
// Requirements honored by all call sites: M % 32 == 0, K % 4 == 0,
// lda/ldb even, base pointers 8-byte aligned (float2 loads).
// N edges: load addresses are CLAMPED (not zeroed) — WMMA output column n
// depends only on B column n, and columns with n >= N are never stored,
// so clamped-garbage is harmless and the inner loop stays branch-free.
// A-layout per ISA: lanes 0-15 hold M=lane rows with K={k,k+1},
// lanes 16-31 the same M rows with K={k+2,k+3}. B symmetric over N.
// C/D: VGPR r -> M = r + 8*(lane>>4), N = lane&15.
// =====================================================================
template <bool TRANS_B>
__global__ void gemm_wmma_f32(const float* __restrict__ A, int lda,
                              const float* __restrict__ Bm, int ldb,
                              const float* __restrict__ bias,
                              float* __restrict__ C, int ldc,
                              int M, int N, int K) {
    const int n0   = blockIdx.x * 32;
    const int m0   = blockIdx.y * 32;
    const int lane = threadIdx.x;      // 0..31 (wave32)
    const int half = lane >> 4;        // 0 or 1
    const int lid  = lane & 15;

    // two A row pointers (M assumed multiple of 32)
    const float* __restrict__ Arow0 = A + (size_t)(m0 + lid) * lda;
    const float* __restrict__ Arow1 = A + (size_t)(m0 + 16 + lid) * lda;

    const int nL = n0 + lid;          // left 16-wide tile column
    const int nR = n0 + 16 + lid;     // right 16-wide tile column
    const int nLc = (nL < N) ? nL : (N - 1);  // clamped safe load index
    const int nRc = (nR < N) ? nR : (N - 1);

    const float* __restrict__ BrowL = TRANS_B ? (Bm + (size_t)nLc * ldb) : nullptr;
    const float* __restrict__ BrowR = TRANS_B ? (Bm + (size_t)nRc * ldb) : nullptr;

    v8f acc00 = {}, acc01 = {}, acc10 = {}, acc11 = {};
    for (int k = 0; k < K; k += 4) {
        const int ka = k + 2 * half;
        const v2f a0 = *(const v2f*)(Arow0 + ka);
        const v2f a1 = *(const v2f*)(Arow1 + ka);
        v2f b0, b1;
        if (TRANS_B) {
            b0 = *(const v2f*)(BrowL + ka);
            b1 = *(const v2f*)(BrowR + ka);
        } else {
            const float* r0 = Bm + (size_t)ka * ldb;
            const float* r1 = r0 + ldb;
            b0.x = r0[nLc]; b0.y = r1[nLc];
            b1.x = r0[nRc]; b1.y = r1[nRc];
        }
        acc00 = __builtin_amdgcn_wmma_f32_16x16x4_f32(
            false, a0, false, b0, (short)0, acc00, false, false);
        acc01 = __builtin_amdgcn_wmma_f32_16x16x4_f32(
            false, a0, false, b1, (short)0, acc01, false, false);
        acc10 = __builtin_amdgcn_wmma_f32_16x16x4_f32(
            false, a1, false, b0, (short)0, acc10, false, false);
        acc11 = __builtin_amdgcn_wmma_f32_16x16x4_f32(
            false, a1, false, b1, (short)0, acc11, false, false);
    }

    const int mBase = m0 + 8 * half;   // rows for acc VGPR r: mBase + r (+16 for acc1x)
    if (nL < N) {
        const float bv = bias ? bias[nL] : 0.0f;
#pragma unroll
        for (int r = 0; r < 8; ++r) {
            C[(size_t)(mBase + r) * ldc + nL]      = acc00[r] + bv;
            C[(size_t)(mBase + 16 + r) * ldc + nL] = acc10[r] + bv;
        }
    }
    if (nR < N) {
        const float bv = bias ? bias[nR] : 0.0f;
#pragma unroll
        for (int r = 0; r < 8; ++r) {
            C[(size_t)(mBase + r) * ldc + nR]      = acc01[r] + bv;
            C[(size_t)(mBase + 16 + r) * ldc + nR] = acc11[r] + bv;
        }
    }
}

// ---------------- small fused kernels ----------------

__global__ void fill_val(float* p, int n, float v) {
    int i = blockIdx.x * blockDim.x + threadIdx.x;
    if (i < n) p[i] = v;
}

__global__ void copy_rows(const float* __restrict__ src, int srs,
                          float* __restrict__ dst, int drs, int rows, int cols) {
    int i = blockIdx.x * blockDim.x + threadIdx.x;
    if (i >= rows * cols) return;
    int r = i / cols, c = i % cols;
    dst[(size_t)r * drs + c] = src[(size_t)r * srs + c];
}

// dst[r, 0:E] = emb[ids[r*stride + offset], :]
__global__ void gather_emb(const float* __restrict__ emb,
                           const int* __restrict__ ids, int stride, int offset,
                           float* __restrict__ dst, int drs, int rows) {
    int i = blockIdx.x * blockDim.x + threadIdx.x;
    if (i >= rows * cE) return;
    int r = i / cE, e = i % cE;
    dst[(size_t)r * drs + e] = emb[(size_t)ids[r * stride + offset] * cE + e];
}

// enc_emb time-major (S,B,E): row (t*B+b) = emb_w[text_ids[b, t+1]]
__global__ void gather_enc(const float* __restrict__ emb,
                           const int* __restrict__ ids, float* __restrict__ dst) {
    int i = blockIdx.x * blockDim.x + threadIdx.x;
    if (i >= cS * cB * cE) return;
    int e = i % cE, r = i / cE;
    int b = r % cB, t = r / cB;
    dst[i] = emb[(size_t)ids[b * cT + t + 1] * cE + e];
}

__global__ void init_w(const int* __restrict__ ids, int* __restrict__ w) {
    int b = blockIdx.x * blockDim.x + threadIdx.x;
    if (b < cB) w[b] = ids[b * cT]; // text_ids[:, 0]
}

// t1[b,h] = lab * lc1_w[h] + lc1_b[h], lab optionally inverted (1-lab)
__global__ void lconn_t1(const int* __restrict__ labels, int invert,
                         const float* __restrict__ lc1_w,
                         const float* __restrict__ lc1_b, float* __restrict__ t1) {
    int i = blockIdx.x * blockDim.x + threadIdx.x;
    if (i >= cB * cH) return;
    int b = i / cH, h = i % cH;
    float lab = (float)labels[b];
    if (invert) lab = 1.0f - lab;
    t1[i] = lab * lc1_w[h] + lc1_b[h];
}

// h_lm0: first (H-300) cols = 1-lab, last 300 = hidden2
__global__ void hlm0_fill(const int* __restrict__ labels,
                          const float* __restrict__ hidden2, float* __restrict__ h) {
    int i = blockIdx.x * blockDim.x + threadIdx.x;
    if (i >= cB * cH) return;
    int b = i / cH, c = i % cH;
    h[i] = (c < cH - 300) ? (1.0f - (float)labels[b])
                          : hidden2[b * 300 + (c - (cH - 300))];
}

// GRU cell fuse: gi/gh are (B,3H) [r|z|n]; hprev strided; optional 2nd output
__global__ void gru_fuse(const float* __restrict__ gi, const float* __restrict__ gh,
                         const float* __restrict__ hprev, int hprs,
                         float* __restrict__ hout, int hors,
                         float* __restrict__ hout2, int ho2rs) {
    int i = blockIdx.x * blockDim.x + threadIdx.x;
    if (i >= cB * cH) return;
    int b = i / cH, h = i % cH;
    const float* gib = gi + (size_t)b * c3H;
    const float* ghb = gh + (size_t)b * c3H;
    float r = 1.0f / (1.0f + expf(-(gib[h] + ghb[h])));
    float z = 1.0f / (1.0f + expf(-(gib[cH + h] + ghb[cH + h])));
    float n = tanhf(gib[2 * cH + h] + r * ghb[2 * cH + h]);
    float hp = hprev[(size_t)b * hprs + h];
    float out = (1.0f - z) * n + z * hp;
    hout[(size_t)b * hors + h] = out;
    if (hout2) hout2[(size_t)b * ho2rs + h] = out;
}

// Attention: e[s]=v.tanh(hproj[b]+encproj[b,s]); softmax; ctx=sum alpha*enc.
// ctx written to dst1[b*drs1+doff1+h] and optionally dst2.
__global__ void attn_ctx(const float* __restrict__ hproj,
                         const float* __restrict__ encproj,
                         const float* __restrict__ enc_outs,
                         const float* __restrict__ attn_v,
                         float* __restrict__ dst1, int drs1, int doff1,
                         float* __restrict__ dst2, int drs2, int doff2) {
    constexpr int BLK = 128;
    const int b = blockIdx.x, tid = threadIdx.x;
    __shared__ float sm[BLK];
    __shared__ float esh[cS];
    __shared__ float alpha[cS];
    const float* hp = hproj + (size_t)b * cH;
    for (int s = 0; s < cS; ++s) {
        const float* ep = encproj + (size_t)(b * cS + s) * cH;
        float p = 0.0f;
        for (int h = tid; h < cH; h += BLK) p += attn_v[h] * tanhf(hp[h] + ep[h]);
        sm[tid] = p; __syncthreads();
        for (int off = BLK / 2; off > 0; off >>= 1) {
            if (tid < off) sm[tid] += sm[tid + off];
            __syncthreads();
        }
        if (tid == 0) esh[s] = sm[0];
        __syncthreads();
    }
    if (tid == 0) {
        float mx = -INFINITY;
        for (int s = 0; s < cS; ++s) mx = fmaxf(mx, esh[s]);
        float den = 0.0f;
        for (int s = 0; s < cS; ++s) { alpha[s] = expf(esh[s] - mx); den += alpha[s]; }
        float inv = 1.0f / den;
        for (int s = 0; s < cS; ++s) alpha[s] *= inv;
    }
    __syncthreads();
    for (int h = tid; h < cH; h += BLK) {
        float c = 0.0f;
        for (int s = 0; s < cS; ++s)
            c += alpha[s] * enc_outs[(size_t)(b * cS + s) * cH + h];
        dst1[(size_t)b * drs1 + doff1 + h] = c;
        if (dst2) dst2[(size_t)b * drs2 + doff2 + h] = c;
    }
}

__global__ void log_softmax_row(const float* __restrict__ x, float* __restrict__ y, int N) {
    constexpr int BLK = 256;
    const int b = blockIdx.x, tid = threadIdx.x;
    __shared__ float sm[BLK];
    const float* xr = x + (size_t)b * N;
    float mx = -INFINITY;
    for (int v = tid; v < N; v += BLK) mx = fmaxf(mx, xr[v]);
    sm[tid] = mx; __syncthreads();
    for (int off = BLK / 2; off > 0; off >>= 1) {
        if (tid < off) sm[tid] = fmaxf(sm[tid], sm[tid + off]);
        __syncthreads();
    }
    mx = sm[0]; __syncthreads();
    float sum = 0.0f;
    for (int v = tid; v < N; v += BLK) sum += expf(xr[v] - mx);
    sm[tid] = sum; __syncthreads();
    for (int off = BLK / 2; off > 0; off >>= 1) {
        if (tid < off) sm[tid] += sm[tid + off];
        __syncthreads();
    }
    const float lse = mx + logf(sm[0]);
    float* yr = y + (size_t)b * N;
    for (int v = tid; v < N; v += BLK) yr[v] = xr[v] - lse;
}

// soft = softmax(lg + g) (GAMMA=1); w[b] = argmax (lowest index on ties)
__global__ void gumbel_softmax_argmax(const float* __restrict__ lg,
                                      const float* __restrict__ g,
                                      float* __restrict__ soft,
                                      int* __restrict__ w, int N) {
    constexpr int BLK = 256;
    const int b = blockIdx.x, tid = threadIdx.x;
    __shared__ float smv[BLK];
    __shared__ int   smi[BLK];
    const float* lr = lg + (size_t)b * N;
    const float* gr = g  + (size_t)b * N;
    float best = -INFINITY; int bi = 0;
    for (int v = tid; v < N; v += BLK) {
        float y = lr[v] + gr[v];
        if (y > best) { best = y; bi = v; }
    }
    smv[tid] = best; smi[tid] = bi; __syncthreads();
    for (int off = BLK / 2; off > 0; off >>= 1) {
        if (tid < off) {
            if (smv[tid + off] > smv[tid] ||
                (smv[tid + off] == smv[tid] && smi[tid + off] < smi[tid])) {
                smv[tid] = smv[tid + off]; smi[tid] = smi[tid + off];
            }
        }
        __syncthreads();
    }
    const float mx = smv[0];
    if (tid == 0) w[b] = smi[0];
    __syncthreads();
    float sum = 0.0f;
    for (int v = tid; v < N; v += BLK) sum += expf(lr[v] + gr[v] - mx);
    smv[tid] = sum; __syncthreads();
    for (int off = BLK / 2; off > 0; off >>= 1) {
        if (tid < off) smv[tid] += smv[tid + off];
        __syncthreads();
    }
    const float inv = 1.0f / smv[0];
    float* sr = soft + (size_t)b * N;
    for (int v = tid; v < N; v += BLK) sr[v] = expf(lr[v] + gr[v] - mx) * inv;
}

// conv1d VALID + relu + max over positions. x time-major (T,B,E); w (NF,E,KW).
template <int KW>
__global__ void conv_relu_maxpool(const float* __restrict__ x_tm,
                                  const float* __restrict__ w,
                                  const float* __restrict__ bias,
                                  float* __restrict__ feats, int featOff) {
    const int b = blockIdx.x, f = blockIdx.y, p = threadIdx.x; // 32 threads
    constexpr int P = cT - KW + 1;
    __shared__ float sm[32];
    float v = 0.0f; // relu floor, exact padding value for p >= P
    if (p < P) {
        float s = bias[f];
        const float* wf = w + (size_t)f * cE * KW;
        for (int j = 0; j < KW; ++j) {
            const float* xr = x_tm + (size_t)((p + j) * cB + b) * cE;
            const float* wj = wf + j;
            for (int e = 0; e < cE; ++e) s += xr[e] * wj[e * KW];
        }
        v = fmaxf(s, 0.0f);
    }
    sm[p] = v; __syncthreads();
    for (int off = 16; off > 0; off >>= 1) {
        if (p < off) sm[p] = fmaxf(sm[p], sm[p + off]);
        __syncthreads();
    }
    if (p == 0) feats[b * (3 * cNF) + featOff + f] = sm[0];
}

__global__ void fc_classes(const float* __restrict__ feats,
                           const float* __restrict__ fc_w,
                           const float* __restrict__ fc_b,
                           float* __restrict__ probs_out,
                           float* __restrict__ classes_out) {
    int b = blockIdx.x * blockDim.x + threadIdx.x;
    if (b >= cB) return;
    float p0 = fc_b[0], p1 = fc_b[1];
    const float* fr = feats + (size_t)b * 3 * cNF;
    for (int j = 0; j < 3 * cNF; ++j) {
        p0 += fr[j] * fc_w[j];
        p1 += fr[j] * fc_w[3 * cNF + j];
    }
    probs_out[b * 2] = p0;
    probs_out[b * 2 + 1] = p1;
    classes_out[b] = (p1 > p0) ? 1.0f : 0.0f;
}

// =====================================================================
// Host orchestration
// =====================================================================
extern "C" void kernel_launch(void* const* d_in, const int* in_sizes, int n_in,
                              void* d_out, int out_size, void* d_ws, size_t ws_size,
                              hipStream_t stream) {
    (void)in_sizes; (void)n_in; (void)out_size; (void)ws_size;
    // ---- inputs (setup_inputs order) ----
    const int*   text_ids = (const int*)  d_in[0];
    const int*   labels   = (const int*)  d_in[1];
    const float* hidden2  = (const float*)d_in[2];
    const float* gnoise   = (const float*)d_in[3];
    const float* emb_w    = (const float*)d_in[4];
    const float* clas_emb = (const float*)d_in[5];
    const float* lm_emb   = (const float*)d_in[6];
    const float* enc_Wih  = (const float*)d_in[7];
    const float* enc_Whh  = (const float*)d_in[8];
    const float* enc_bih  = (const float*)d_in[9];
    const float* enc_bhh  = (const float*)d_in[10];
    const float* lc1_w    = (const float*)d_in[11];
    const float* lc1_b    = (const float*)d_in[12];
    const float* lc2_w    = (const float*)d_in[13];
    const float* lc2_b    = (const float*)d_in[14];
    const float* conn_w   = (const float*)d_in[15];
    const float* conn_b   = (const float*)d_in[16];
    const float* dec_Wih  = (const float*)d_in[17];
    const float* dec_Whh  = (const float*)d_in[18];
    const float* dec_bih  = (const float*)d_in[19];
    const float* dec_bhh  = (const float*)d_in[20];
    const float* attn_Wa  = (const float*)d_in[21];
    const float* attn_v   = (const float*)d_in[22];
    const float* out_w    = (const float*)d_in[23];
    const float* out_b    = (const float*)d_in[24];
    const float* conv3_w  = (const float*)d_in[25];
    const float* conv3_b  = (const float*)d_in[26];
    const float* conv4_w  = (const float*)d_in[27];
    const float* conv4_b  = (const float*)d_in[28];
    const float* conv5_w  = (const float*)d_in[29];
    const float* conv5_b  = (const float*)d_in[30];
    const float* fc_w     = (const float*)d_in[31];
    const float* fc_b     = (const float*)d_in[32];
    const float* lm_Wih   = (const float*)d_in[33];
    const float* lm_Whh   = (const float*)d_in[34];
    const float* lm_bih   = (const float*)d_in[35];
    const float* lm_bhh   = (const float*)d_in[36];
    const float* lmout_w  = (const float*)d_in[37];
    const float* lmout_b  = (const float*)d_in[38];

    // ---- outputs (tuple order, flat) ----
    float* out_dec     = (float*)d_out;                       // (T,B,V) log-probs
    float* out_soft    = out_dec  + (size_t)cT * cB * cV;     // (T,B,V)
    float* out_probs   = out_soft + (size_t)cT * cB * cV;     // (B,2)
    float* out_classes = out_probs + (size_t)cB * 2;          // (B,)
    float* out_lm      = out_classes + cB;                    // (T,B,V)

    // ---- workspace carve ----
    float* wsf = (float*)d_ws;
    size_t off = 0;
    auto wsalloc = [&](size_t n) { float* p = wsf + off; off += n; return p; };
    float* enc_emb  = wsalloc((size_t)cS * cB * cE);
    float* enc_outs = wsalloc((size_t)cB * cS * cH);   // (B,S,H)
    float* encproj  = wsalloc((size_t)cB * cS * cH);   // enc @ Wa_enc^T (precomputed)
    float* hbuf     = wsalloc((size_t)cB * cH);
    float* gi       = wsalloc((size_t)cB * c3H);
    float* gh       = wsalloc((size_t)cB * c3H);
    float* t1       = wsalloc((size_t)cB * cH);
    float* cc       = wsalloc((size_t)cB * cH);        // [lconn | z]
    float* oc       = wsalloc((size_t)cB * 2 * cH);    // [h | ctx] TF decoder
    float* ocg      = wsalloc((size_t)cB * 2 * cH);    // [h | ctx] gumbel decoder
    float* xin      = wsalloc((size_t)cB * (cE + cH)); // [emb | ctx] TF
    float* xing     = wsalloc((size_t)cB * (cE + cH)); // gumbel
    float* hproj    = wsalloc((size_t)cB * cH);
    float* lgb      = wsalloc((size_t)cB * cV);
    float* lmin_tm  = wsalloc((size_t)cT * cB * cE);
    float* clas_tm  = wsalloc((size_t)cT * cB * cE);
    float* lmh_tm   = wsalloc((size_t)cT * cB * cH);
    float* hlm0     = wsalloc((size_t)cB * cH);
    float* xlm0     = wsalloc((size_t)cB * cE);
    float* feats    = wsalloc((size_t)cB * 3 * cNF);
    int*   wbuf     = (int*)wsalloc(cB);

    auto cdiv = [](int a, int b) { return (a + b - 1) / b; };
    auto gemm_tn = [&](const float* A, int lda, const float* W, int ldb,
                       const float* bias, float* C, int ldc, int M, int N, int K) {
        dim3 g(cdiv(N, 32), cdiv(M, 32));
        gemm_wmma_f32<true><<<g, 32, 0, stream>>>(A, lda, W, ldb, bias, C, ldc, M, N, K);
    };
    auto gemm_nn = [&](const float* A, int lda, const float* Bm, int ldb,
                       const float* bias, float* C, int ldc, int M, int N, int K) {
        dim3 g(cdiv(N, 32), cdiv(M, 32));
        gemm_wmma_f32<false><<<g, 32, 0, stream>>>(A, lda, Bm, ldb, bias, C, ldc, M, N, K);
    };

    // ============ 1) Encoder GRU ============
    gather_enc<<<cdiv(cS * cB * cE, 256), 256, 0, stream>>>(emb_w, text_ids, enc_emb);
    fill_val<<<cdiv(cB * cH, 256), 256, 0, stream>>>(hbuf, cB * cH, 0.0f);
    for (int t = 0; t < cS; ++t) {
        gemm_tn(enc_emb + (size_t)t * cB * cE, cE, enc_Wih, cE, enc_bih, gi, c3H, cB, c3H, cE);
        gemm_tn(hbuf, cH, enc_Whh, cH, enc_bhh, gh, c3H, cB, c3H, cH);
        gru_fuse<<<cdiv(cB * cH, 256), 256, 0, stream>>>(
            gi, gh, hbuf, cH, hbuf, cH, enc_outs + (size_t)t * cH, cS * cH);
    }
    // enc_proj = enc_outs @ Wa_enc^T  (Wa_enc = attn_Wa[:, H:], ldb = 2H) — hoisted
    gemm_tn(enc_outs, cH, attn_Wa + cH, 2 * cH, nullptr, encproj, cH, cB * cS, cH, cH);

    // ============ 2) Initial decoder hiddens ============
    lconn_t1<<<cdiv(cB * cH, 256), 256, 0, stream>>>(labels, 0, lc1_w, lc1_b, t1);
    gemm_tn(t1, cH, lc2_w, cH, lc2_b, cc, cH, cB, cDC, cH);
    copy_rows<<<cdiv(cB * (cH - cDC), 256), 256, 0, stream>>>(
        hbuf + cDC, cH, cc + cDC, cH, cB, cH - cDC); // z = final[:, DC:]
    gemm_tn(cc, cH, conn_w, cH, conn_b, oc, 2 * cH, cB, cH, cH);
    lconn_t1<<<cdiv(cB * cH, 256), 256, 0, stream>>>(labels, 1, lc1_w, lc1_b, t1);
    gemm_tn(t1, cH, lc2_w, cH, lc2_b, cc, cH, cB, cDC, cH);
    gemm_tn(cc, cH, conn_w, cH, conn_b, ocg, 2 * cH, cB, cH, cH);

    init_w<<<1, cB, 0, stream>>>(text_ids, wbuf);

    // ============ 3) Teacher-forced decoder ============
    for (int t = 0; t < cT; ++t) {
        gather_emb<<<cdiv(cB * cE, 256), 256, 0, stream>>>(
            emb_w, text_ids, cT, t, xin, cE + cH, cB);
        gemm_tn(oc, 2 * cH, attn_Wa, 2 * cH, nullptr, hproj, cH, cB, cH, cH);
        attn_ctx<<<cB, 128, 0, stream>>>(hproj, encproj, enc_outs, attn_v,
                                         xin, cE + cH, cE, oc, 2 * cH, cH);
        gemm_tn(xin, cE + cH, dec_Wih, cE + cH, dec_bih, gi, c3H, cB, c3H, cE + cH);
        gemm_tn(oc, 2 * cH, dec_Whh, cH, dec_bhh, gh, c3H, cB, c3H, cH);
        gru_fuse<<<cdiv(cB * cH, 256), 256, 0, stream>>>(
            gi, gh, oc, 2 * cH, oc, 2 * cH, nullptr, 0);
        gemm_tn(oc, 2 * cH, out_w, 2 * cH, out_b, lgb, cV, cB, cV, 2 * cH);
        log_softmax_row<<<cB, 256, 0, stream>>>(lgb, out_dec + (size_t)t * cB * cV, cV);
    }

    // ============ 4) Gumbel decoder ============
    for (int t = 0; t < cT; ++t) {
        gather_emb<<<cdiv(cB * cE, 256), 256, 0, stream>>>(
            emb_w, wbuf, 1, 0, xing, cE + cH, cB);
        gemm_tn(ocg, 2 * cH, attn_Wa, 2 * cH, nullptr, hproj, cH, cB, cH, cH);
        attn_ctx<<<cB, 128, 0, stream>>>(hproj, encproj, enc_outs, attn_v,
                                         xing, cE + cH, cE, ocg, 2 * cH, cH);
        gemm_tn(xing, cE + cH, dec_Wih, cE + cH, dec_bih, gi, c3H, cB, c3H, cE + cH);
        gemm_tn(ocg, 2 * cH, dec_Whh, cH, dec_bhh, gh, c3H, cB, c3H, cH);
        gru_fuse<<<cdiv(cB * cH, 256), 256, 0, stream>>>(
            gi, gh, ocg, 2 * cH, ocg, 2 * cH, nullptr, 0);
        gemm_tn(ocg, 2 * cH, out_w, 2 * cH, out_b, lgb, cV, cB, cV, 2 * cH);
        gumbel_softmax_argmax<<<cB, 256, 0, stream>>>(
            lgb, gnoise + (size_t)t * cB * cV, out_soft + (size_t)t * cB * cV, wbuf, cV);
    }

    // ============ 5) Classifier ============
    gemm_nn(out_soft, cV, clas_emb, cE, nullptr, clas_tm, cE, cT * cB, cE, cV);
    {
        dim3 g(cB, cNF);
        conv_relu_maxpool<3><<<g, 32, 0, stream>>>(clas_tm, conv3_w, conv3_b, feats, 0);
        conv_relu_maxpool<4><<<g, 32, 0, stream>>>(clas_tm, conv4_w, conv4_b, feats, cNF);
        conv_relu_maxpool<5><<<g, 32, 0, stream>>>(clas_tm, conv5_w, conv5_b, feats, 2 * cNF);
    }
    fc_classes<<<1, cB, 0, stream>>>(feats, fc_w, fc_b, out_probs, out_classes);

    // ============ 6) Language model ============
    hlm0_fill<<<cdiv(cB * cH, 256), 256, 0, stream>>>(labels, hidden2, hlm0);
    gather_emb<<<cdiv(cB * cE, 256), 256, 0, stream>>>(
        lm_emb, text_ids, cT, 0, xlm0, cE, cB);
    gemm_nn(out_soft, cV, lm_emb, cE, nullptr, lmin_tm, cE, cT * cB, cE, cV);
    for (int t = 0; t < cT; ++t) {
        const float* x  = (t == 0) ? xlm0 : lmin_tm + (size_t)(t - 1) * cB * cE;
        const float* hp = (t == 0) ? hlm0 : lmh_tm + (size_t)(t - 1) * cB * cH;
        gemm_tn(x, cE, lm_Wih, cE, lm_bih, gi, c3H, cB, c3H, cE);
        gemm_tn(hp, cH, lm_Whh, cH, lm_bhh, gh, c3H, cB, c3H, cH);
        gru_fuse<<<cdiv(cB * cH, 256), 256, 0, stream>>>(
            gi, gh, hp, cH, lmh_tm + (size_t)t * cB * cH, cH, nullptr, 0);
    }
    gemm_tn(lmh_tm, cH, lmout_w, cH, lmout_b, out_lm, cV, cT * cB, cV, cH);
}